// HGMPPretrainModel_11914239279720
// MI455X (gfx1250) — compile-verified
//
#include <hip/hip_runtime.h>
#include <math.h>
#include <stdint.h>

typedef __attribute__((ext_vector_type(2))) float v2f;
typedef __attribute__((ext_vector_type(8))) float v8f;
typedef __attribute__((ext_vector_type(4))) int v4i;

#define NPAPER 300000
#define NAUTHOR 150000
#define E_CITES 600000
#define E_WRITES 300000
#define E_WRITTEN 300000
#define HIDDEN 128
#define GSEG 1024
#define KCHUNK 64
#define ASTRIDE (KCHUNK + 4)  // padded LDS row stride: banks spread, 16B aligned

#if __has_builtin(__builtin_amdgcn_global_load_async_to_lds_b128) && \
    __has_builtin(__builtin_amdgcn_s_wait_asynccnt)
#define ASYNC_STAGE 1
#else
#define ASYNC_STAGE 0
#endif

enum { ACT_NONE = 0, ACT_RELU = 1, ACT_GELU = 2 };

__device__ __forceinline__ float gelu_exact(float x) {
  return 0.5f * x * (1.0f + erff(x * 0.70710678118654752440f));
}

// ---------------------------------------------------------------------------
// WMMA fp32 GEMM:  C[nrows x NCOL] = act(A)[nrows x K] * B[K x NCOL] (+ bias)
// 256 threads = 8 wave32s; each wave computes a 16 x NCOL stripe with
// v_wmma_f32_16x16x4_f32.
//  - W chunk staged into LDS k-pair-interleaved so each B fragment is one
//    aligned ds_load_b64 (no repack movs in the WMMA loop).
//  - All NT B-fragments of a k-step are fetched into registers before the
//    WMMA burst so LDS latency overlaps matrix issue within a wave.
//  - A chunk staged via GLOBAL_LOAD_ASYNC_TO_LDS_B128 (ASYNCcnt) when no
//    input activation is fused; padded row stride avoids bank conflicts.
// ---------------------------------------------------------------------------
template <int ACT, int NCOL>
__global__ __launch_bounds__(256) void wmma_gemm(
    const float* __restrict__ A, int lda,
    const float* __restrict__ B, int ldb,
    const float* __restrict__ bias,
    float* __restrict__ C, int ldc,
    int nrows, int K) {
  constexpr int NT = NCOL / 16;
  __shared__ __align__(16) float lds_a[128 * ASTRIDE];
  __shared__ __align__(16) float lds_w[KCHUNK * NCOL];  // [(k/2)][n][2] pairs

  const int tid = threadIdx.x;
  const int lane = tid & 31;
  const int wave = tid >> 5;
  const int hl = lane & 15;
  const int koff = (lane >= 16) ? 2 : 0;
  const int row0b = blockIdx.x * 128;

  v8f acc[NT];
#pragma unroll
  for (int nt = 0; nt < NT; ++nt) {
    float b0 = bias ? bias[nt * 16 + hl] : 0.0f;
#pragma unroll
    for (int i = 0; i < 8; ++i) acc[nt][i] = b0;
  }

  for (int kc = 0; kc < K; kc += KCHUNK) {
    // Stage B chunk (KCHUNK x NCOL) into LDS, k-pair interleaved:
    // element (k, n) -> lds_w[((k>>1)*NCOL + n)*2 + (k&1)]
#pragma unroll 1
    for (int q = tid * 4; q < KCHUNK * NCOL; q += 1024) {
      int kr = q / NCOL, cc = q % NCOL;
      float4 v = *(const float4*)&B[(size_t)(kc + kr) * ldb + cc];
      float* dst = &lds_w[(size_t)(((kr >> 1) * NCOL + cc) * 2 + (kr & 1))];
      dst[0] = v.x; dst[2] = v.y; dst[4] = v.z; dst[6] = v.w;
    }
    // Stage A chunk (128 rows x KCHUNK) into LDS.
#if ASYNC_STAGE
    if (ACT == ACT_NONE) {
#pragma unroll 1
      for (int q = tid * 4; q < 128 * KCHUNK; q += 1024) {
        int r = q / KCHUNK, cc = q % KCHUNK;
        int row = row0b + r;
        if (row >= nrows) row = nrows - 1;  // clamp; tail masked at store
        const float* g = &A[(size_t)row * lda + kc + cc];
        float* l = &lds_a[r * ASTRIDE + cc];
        __builtin_amdgcn_global_load_async_to_lds_b128(
            (v4i*)(uintptr_t)g, (v4i*)l, 0, 0);
      }
      __builtin_amdgcn_s_wait_asynccnt(0);
    } else
#endif
    {
#pragma unroll 1
      for (int q = tid * 4; q < 128 * KCHUNK; q += 1024) {
        int r = q / KCHUNK, cc = q % KCHUNK;
        int row = row0b + r;
        if (row >= nrows) row = nrows - 1;
        float4 v = *(const float4*)&A[(size_t)row * lda + kc + cc];
        if (ACT == ACT_RELU) {
          v.x = fmaxf(v.x, 0.f); v.y = fmaxf(v.y, 0.f);
          v.z = fmaxf(v.z, 0.f); v.w = fmaxf(v.w, 0.f);
        } else if (ACT == ACT_GELU) {
          v.x = gelu_exact(v.x); v.y = gelu_exact(v.y);
          v.z = gelu_exact(v.z); v.w = gelu_exact(v.w);
        }
        *(float4*)&lds_a[r * ASTRIDE + cc] = v;
      }
    }
    if (kc + KCHUNK < K)  // speculative prefetch of the next W chunk
      __builtin_prefetch(&B[(size_t)(kc + KCHUNK) * ldb + tid * 4], 0, 0);
    __syncthreads();

    const int ar = (wave * 16 + hl) * ASTRIDE;
    const v2f* wpair = (const v2f*)lds_w;
#pragma unroll
    for (int k = 0; k < KCHUNK; k += 4) {
      v2f a;
      a.x = lds_a[ar + k + koff];
      a.y = lds_a[ar + k + koff + 1];
      const int pb = ((k + koff) >> 1) * NCOL + hl;
      v2f bf[NT];
#pragma unroll
      for (int nt = 0; nt < NT; ++nt) bf[nt] = wpair[pb + nt * 16];
#pragma unroll
      for (int nt = 0; nt < NT; ++nt) {
        acc[nt] = __builtin_amdgcn_wmma_f32_16x16x4_f32(
            false, a, false, bf[nt], (short)0, acc[nt], false, false);
      }
    }
    __syncthreads();
  }

  const int rbase = row0b + wave * 16 + ((lane >= 16) ? 8 : 0);
#pragma unroll
  for (int nt = 0; nt < NT; ++nt) {
    int cn = nt * 16 + hl;
#pragma unroll
    for (int r = 0; r < 8; ++r) {
      int row = rbase + r;
      if (row < nrows) C[(size_t)row * ldc + cn] = acc[nt][r];
    }
  }
}

// ---------------------------------------------------------------------------
// Order-preserving float<->uint encoding for atomic float max.
// ---------------------------------------------------------------------------
__device__ __forceinline__ unsigned enc_f(float f) {
  unsigned u = __float_as_uint(f);
  return (u & 0x80000000u) ? ~u : (u | 0x80000000u);
}
__device__ __forceinline__ float dec_f(unsigned u) {
  return __uint_as_float((u & 0x80000000u) ? (u ^ 0x80000000u) : ~u);
}

__global__ void fill_f32(float* __restrict__ p, float v, long n) {
  long i = (long)blockIdx.x * blockDim.x + threadIdx.x;
  if (i < n) p[i] = v;
}

// Pass 1: alpha[e,h] = (q[dst] . ke[src]) * p_rel[h] / sqrt(DH); seg-max.
__global__ void edge_alpha(const int* __restrict__ es, const int* __restrict__ ed,
                           int E, const float* __restrict__ q,
                           const float* __restrict__ ke,
                           const float* __restrict__ p_rel,
                           float* __restrict__ alpha,
                           unsigned* __restrict__ amax) {
  long idx = (long)blockIdx.x * blockDim.x + threadIdx.x;
  if (idx >= (long)E * 2) return;
  int e = (int)(idx >> 1), h = (int)(idx & 1);
  int s = es[e], d = ed[e];
  const float4* qa = (const float4*)(q + (size_t)d * HIDDEN + h * 64);
  const float4* ka = (const float4*)(ke + (size_t)s * HIDDEN + h * 64);
  float acc = 0.f;
#pragma unroll
  for (int i = 0; i < 16; ++i) {
    float4 a = qa[i], b = ka[i];
    acc += a.x * b.x + a.y * b.y + a.z * b.z + a.w * b.w;
  }
  float al = acc * p_rel[h] * 0.125f;  // 1/sqrt(64)
  alpha[idx] = al;
  atomicMax(&amax[(size_t)d * 2 + h], enc_f(al));
}

// Pass 2: ex = exp(alpha - amax[dst]); seg-sum into den.
__global__ void edge_exp(const int* __restrict__ ed, int E,
                         float* __restrict__ alpha,
                         const unsigned* __restrict__ amax,
                         float* __restrict__ den) {
  long idx = (long)blockIdx.x * blockDim.x + threadIdx.x;
  if (idx >= (long)E * 2) return;
  int e = (int)(idx >> 1), h = (int)(idx & 1);
  int d = ed[e];
  float m = dec_f(amax[(size_t)d * 2 + h]);
  float ex = expf(alpha[idx] - m);
  alpha[idx] = ex;
  atomicAdd(&den[(size_t)d * 2 + h], ex);
}

// Pass 3: agg[dst] += ve[src] * (ex / den[dst])  (128 floats per edge).
__global__ void edge_scatter(const int* __restrict__ es, const int* __restrict__ ed,
                             int E, const float* __restrict__ ve,
                             const float* __restrict__ alpha,
                             const float* __restrict__ den,
                             float* __restrict__ agg) {
  long idx = (long)blockIdx.x * blockDim.x + threadIdx.x;
  if (idx >= (long)E * HIDDEN) return;
  int e = (int)(idx >> 7), c = (int)(idx & 127), h = c >> 6;
  int s = es[e], d = ed[e];
  float w = alpha[(size_t)e * 2 + h] / den[(size_t)d * 2 + h];
  atomicAdd(&agg[(size_t)d * HIDDEN + c], ve[(size_t)s * HIDDEN + c] * w);
}

// h = relu(sigmoid(skip)*o + (1-sigmoid(skip))*h)
__global__ void combine_relu(float* __restrict__ h, const float* __restrict__ o,
                             const float* __restrict__ skip, long n) {
  long i = (long)blockIdx.x * blockDim.x + threadIdx.x;
  if (i >= n) return;
  float a = 1.0f / (1.0f + expf(-skip[0]));
  float v = a * o[i] + (1.0f - a) * h[i];
  h[i] = fmaxf(v, 0.0f);
}

__global__ void pool_accum(const float* __restrict__ h, const int* __restrict__ batch,
                           int n, float* __restrict__ sums, float* __restrict__ cnt) {
  long idx = (long)blockIdx.x * blockDim.x + threadIdx.x;
  if (idx >= (long)n * HIDDEN) return;
  int node = (int)(idx >> 7), c = (int)(idx & 127);
  int b = batch[node];
  atomicAdd(&sums[(size_t)b * HIDDEN + c], h[idx]);
  if (c == 0) atomicAdd(&cnt[b], 1.0f);
}

__global__ void pool_norm_add(float* __restrict__ pooled, const float* __restrict__ sums,
                              const float* __restrict__ cnt) {
  long idx = (long)blockIdx.x * blockDim.x + threadIdx.x;
  if (idx >= (long)GSEG * HIDDEN) return;
  int b = (int)(idx >> 7);
  float c = fmaxf(cnt[b], 1.0f);
  pooled[idx] += sums[idx] / c;
}

// ---------------------------------------------------------------------------
// Host orchestration
// ---------------------------------------------------------------------------
static inline dim3 g1(long n, int b) { return dim3((unsigned)((n + b - 1) / b)); }

static void launch_gemm(hipStream_t st, const float* A, int lda, const float* B, int ldb,
                        const float* bias, float* C, int ldc, int nrows, int K,
                        int ncol, int act) {
  dim3 grid((nrows + 127) / 128), blk(256);
  if (ncol == 128) {
    if (act == ACT_NONE)
      wmma_gemm<ACT_NONE, 128><<<grid, blk, 0, st>>>(A, lda, B, ldb, bias, C, ldc, nrows, K);
    else if (act == ACT_RELU)
      wmma_gemm<ACT_RELU, 128><<<grid, blk, 0, st>>>(A, lda, B, ldb, bias, C, ldc, nrows, K);
    else
      wmma_gemm<ACT_GELU, 128><<<grid, blk, 0, st>>>(A, lda, B, ldb, bias, C, ldc, nrows, K);
  } else {
    wmma_gemm<ACT_NONE, 64><<<grid, blk, 0, st>>>(A, lda, B, ldb, bias, C, ldc, nrows, K);
  }
}

static void run_relation(hipStream_t st, const int* eidx, int E,
                         const float* a_rel, const float* m_rel, const float* p_rel,
                         const float* k_src, const float* v_src, int ns,
                         const float* q_dst, float* agg_dst, int nd,
                         float* keS, float* veS, float* alpha, unsigned* amaxu,
                         float* den) {
  // Relation-specific keys/values: per-head 64x64 GEMMs over all source nodes.
  for (int h = 0; h < 2; ++h) {
    launch_gemm(st, k_src + h * 64, HIDDEN, a_rel + h * 64 * 64, 64, nullptr,
                keS + h * 64, HIDDEN, ns, 64, 64, ACT_NONE);
    launch_gemm(st, v_src + h * 64, HIDDEN, m_rel + h * 64 * 64, 64, nullptr,
                veS + h * 64, HIDDEN, ns, 64, 64, ACT_NONE);
  }
  fill_f32<<<g1((long)nd * 2, 256), 256, 0, st>>>((float*)amaxu, 0.0f, (long)nd * 2);
  fill_f32<<<g1((long)nd * 2, 256), 256, 0, st>>>(den, 0.0f, (long)nd * 2);
  const int* es = eidx;
  const int* ed = eidx + E;
  edge_alpha<<<g1((long)E * 2, 256), 256, 0, st>>>(es, ed, E, q_dst, keS, p_rel, alpha, amaxu);
  edge_exp<<<g1((long)E * 2, 256), 256, 0, st>>>(ed, E, alpha, amaxu, den);
  edge_scatter<<<g1((long)E * HIDDEN, 256), 256, 0, st>>>(es, ed, E, veS, alpha, den, agg_dst);
}

extern "C" void kernel_launch(void* const* d_in, const int* in_sizes, int n_in,
                              void* d_out, int out_size, void* d_ws, size_t ws_size,
                              hipStream_t stream) {
  (void)in_sizes; (void)n_in; (void)out_size; (void)ws_size;
  hipStream_t st = stream;
#define FIN(i) ((const float*)d_in[(i)])
#define IIN(i) ((const int*)d_in[(i)])

  const float* x_paper = FIN(0);
  const float* x_author = FIN(1);
  const int* e_cites = IIN(2);
  const int* e_writes = IIN(3);
  const int* e_written = IIN(4);
  const int* batch_paper = IIN(5);
  const int* batch_author = IIN(6);
  // params pytree (jax sorted-key flattening): adapter -> layers -> proj
  const float* ad_a_b = FIN(7);
  const float* ad_a_w = FIN(8);
  const float* ad_p_b = FIN(9);
  const float* ad_p_w = FIN(10);
  // layers start at 11, 27 leaves each (see offsets in layer loop)
  const float* pj_l1_b = FIN(65);
  const float* pj_l1_w = FIN(66);
  const float* pj_l2_b = FIN(67);
  const float* pj_l2_w = FIN(68);

  // Workspace carve-up (fp32).
  float* ws = (float*)d_ws;
  size_t off = 0;
  auto alloc = [&](size_t n) { float* p = ws + off; off += n; return p; };
  const size_t NPH = (size_t)NPAPER * HIDDEN, NAH = (size_t)NAUTHOR * HIDDEN;
  float* hP = alloc(NPH);
  float* hA = alloc(NAH);
  float* kP = alloc(NPH); float* qP = alloc(NPH); float* vP = alloc(NPH);
  float* kA = alloc(NAH); float* qA = alloc(NAH); float* vA = alloc(NAH);
  float* keS = alloc(NPH);
  float* veS = alloc(NPH);
  float* aggP = alloc(NPH);
  float* aggA = alloc(NAH);
  float* tmp = alloc(NPH);
  float* alpha = alloc((size_t)E_CITES * 2);
  unsigned* amaxu = (unsigned*)alloc((size_t)NPAPER * 2);
  float* den = alloc((size_t)NPAPER * 2);
  float* pooled = alloc((size_t)GSEG * HIDDEN);
  float* sums = alloc((size_t)GSEG * HIDDEN);
  float* cnt = alloc((size_t)GSEG);

  // Feature adapters.
  launch_gemm(st, x_paper, 256, ad_p_w, 128, ad_p_b, hP, 128, NPAPER, 256, 128, ACT_NONE);
  launch_gemm(st, x_author, 64, ad_a_w, 128, ad_a_b, hA, 128, NAUTHOR, 64, 128, ACT_NONE);

  for (int L = 0; L < 2; ++L) {
    int B0 = 11 + L * 27;
    // +0 a_lin.author.b +1 a_lin.author.w +2 a_lin.paper.b +3 a_lin.paper.w
    // +4..6 a_rel{cites,writes,written_by}
    // +7 k.author.b +8 k.author.w +9 k.paper.b +10 k.paper.w
    // +11..13 m_rel{...} +14..16 p_rel{...}
    // +17 q.author.b +18 q.author.w +19 q.paper.b +20 q.paper.w
    // +21 skip.author +22 skip.paper
    // +23 v.author.b +24 v.author.w +25 v.paper.b +26 v.paper.w
    launch_gemm(st, hP, 128, FIN(B0 + 10), 128, FIN(B0 + 9), kP, 128, NPAPER, 128, 128, ACT_NONE);
    launch_gemm(st, hP, 128, FIN(B0 + 20), 128, FIN(B0 + 19), qP, 128, NPAPER, 128, 128, ACT_NONE);
    launch_gemm(st, hP, 128, FIN(B0 + 26), 128, FIN(B0 + 25), vP, 128, NPAPER, 128, 128, ACT_NONE);
    launch_gemm(st, hA, 128, FIN(B0 + 8), 128, FIN(B0 + 7), kA, 128, NAUTHOR, 128, 128, ACT_NONE);
    launch_gemm(st, hA, 128, FIN(B0 + 18), 128, FIN(B0 + 17), qA, 128, NAUTHOR, 128, 128, ACT_NONE);
    launch_gemm(st, hA, 128, FIN(B0 + 24), 128, FIN(B0 + 23), vA, 128, NAUTHOR, 128, 128, ACT_NONE);

    fill_f32<<<g1((long)NPH, 256), 256, 0, st>>>(aggP, 0.0f, (long)NPH);
    fill_f32<<<g1((long)NAH, 256), 256, 0, st>>>(aggA, 0.0f, (long)NAH);

    // paper -cites-> paper
    run_relation(st, e_cites, E_CITES, FIN(B0 + 4), FIN(B0 + 11), FIN(B0 + 14),
                 kP, vP, NPAPER, qP, aggP, NPAPER, keS, veS, alpha, amaxu, den);
    // author -writes-> paper
    run_relation(st, e_writes, E_WRITES, FIN(B0 + 5), FIN(B0 + 12), FIN(B0 + 15),
                 kA, vA, NAUTHOR, qP, aggP, NPAPER, keS, veS, alpha, amaxu, den);
    // paper -written_by-> author
    run_relation(st, e_written, E_WRITTEN, FIN(B0 + 6), FIN(B0 + 13), FIN(B0 + 16),
                 kP, vP, NPAPER, qA, aggA, NAUTHOR, keS, veS, alpha, amaxu, den);

    // o = gelu(agg) @ a_lin + b ; h = relu(sig(skip)*o + (1-sig)*h)
    launch_gemm(st, aggP, 128, FIN(B0 + 3), 128, FIN(B0 + 2), tmp, 128, NPAPER, 128, 128, ACT_GELU);
    combine_relu<<<g1((long)NPH, 256), 256, 0, st>>>(hP, tmp, FIN(B0 + 22), (long)NPH);
    launch_gemm(st, aggA, 128, FIN(B0 + 1), 128, FIN(B0 + 0), tmp, 128, NAUTHOR, 128, 128, ACT_GELU);
    combine_relu<<<g1((long)NAH, 256), 256, 0, st>>>(hA, tmp, FIN(B0 + 21), (long)NAH);
  }

  // Readout: per-seed-graph mean pool per node type, summed across types.
  const long GH = (long)GSEG * HIDDEN;
  fill_f32<<<g1(GH, 256), 256, 0, st>>>(pooled, 0.0f, GH);
  fill_f32<<<g1(GH, 256), 256, 0, st>>>(sums, 0.0f, GH);
  fill_f32<<<g1(GSEG, 256), 256, 0, st>>>(cnt, 0.0f, GSEG);
  pool_accum<<<g1((long)NPH, 256), 256, 0, st>>>(hP, batch_paper, NPAPER, sums, cnt);
  pool_norm_add<<<g1(GH, 256), 256, 0, st>>>(pooled, sums, cnt);
  fill_f32<<<g1(GH, 256), 256, 0, st>>>(sums, 0.0f, GH);
  fill_f32<<<g1(GSEG, 256), 256, 0, st>>>(cnt, 0.0f, GSEG);
  pool_accum<<<g1((long)NAH, 256), 256, 0, st>>>(hA, batch_author, NAUTHOR, sums, cnt);
  pool_norm_add<<<g1(GH, 256), 256, 0, st>>>(pooled, sums, cnt);

  // Projection head: l1 -> ReLU (fused into l2's A path) -> l2.
  launch_gemm(st, pooled, 128, pj_l1_w, 128, pj_l1_b, tmp, 128, GSEG, 128, 128, ACT_NONE);
  launch_gemm(st, tmp, 128, pj_l2_w, 128, pj_l2_b, (float*)d_out, 128, GSEG, 128, 128, ACT_RELU);
}